// LSTMNET_64879775973874
// MI455X (gfx1250) — compile-verified
//
#include <hip/hip_runtime.h>
#include <hip/hip_bf16.h>

typedef __attribute__((ext_vector_type(2))) float v2f;
typedef __attribute__((ext_vector_type(8))) float v8f;

// One wave (32 lanes) per 16 batch elements.
// Per step: D[16 gate-rows x 16 batch] = A(W_hh 16x4) x B(h^T 4x16) + C(x-gate term)
// via v_wmma_f32_16x16x4_f32 (pure fp32 -> matches reference numerics).
//
// Gate rows are PERMUTED as [i0,f0,g0,o0, i1,f1,g1,o1, ...]: old_row(r) =
// (r&3)*4 + (r>>2).  With the C/D layout (lane L, VGPR v -> row M = v+8*(L>=16),
// batch N = L&15), each lane holds complete (i,f,g,o) quadruples for hidden dims
// n0 = 2*hi, n1 = 2*hi+1 of its batch: D[0..3] / D[4..7].  Cell update is fully
// lane-local (no shuffles/selects in the loop), and the B operand layout
// (VGPR0: K=0|2 per half, VGPR1: K=1|3) is exactly the two h values each half
// just produced.  A layout: lane L -> row M=L&15, VGPR v -> K = v + 2*(L>=16).
//
// Activations use gfx1250's hardware V_TANH_F32.  Sigmoid rows (type != g) have
// their A row and C constants PRE-SCALED by 0.5, so in-loop:
//   sigmoid slot: act = fma(0.5, tanh(D), 0.5)     (no pre-multiply)
//   g slot:       act = tanh(D)                    (row unscaled)

#if __has_builtin(__builtin_amdgcn_tanhf)
#define HW_TANH(x) __builtin_amdgcn_tanhf(x)
#elif __has_builtin(__builtin_amdgcn_tanh_f32)
#define HW_TANH(x) __builtin_amdgcn_tanh_f32(x)
#endif

__device__ __forceinline__ float fast_tanh(float x) {
#ifdef HW_TANH
    return HW_TANH(x);
#else
    const float L2E = 1.44269504088896340736f;
    return fmaf(2.0f, __builtin_amdgcn_rcpf(1.0f + __builtin_amdgcn_exp2f(-2.0f * L2E * x)), -1.0f);
#endif
}
// input already pre-scaled by 0.5: sigmoid(2y) = 0.5 + 0.5*tanh(y)
__device__ __forceinline__ float sigmoid_of_half(float y) {
    return fmaf(0.5f, fast_tanh(y), 0.5f);
}

__global__ __launch_bounds__(32) void lstm_wmma_f32_16x16x4(
    const float* __restrict__ x,     // [B, T]
    const float* __restrict__ W_ih,  // [16, 1]
    const float* __restrict__ W_hh,  // [16, 4]
    const float* __restrict__ b_ih,  // [16]
    const float* __restrict__ b_hh,  // [16]
    const float* __restrict__ W1,    // [10, 4]
    const float* __restrict__ b1,    // [10]
    const float* __restrict__ Wll,   // [7, 10]
    const float* __restrict__ bll,   // [7]
    float* __restrict__ out,         // [B, 7]
    int T)
{
    const int lane = threadIdx.x;        // 0..31 (wave32)
    const int hi   = lane >> 4;          // 0 = lower half (dims 0,1), 1 = upper (dims 2,3)
    const int m    = lane & 15;
    const int batch = blockIdx.x * 16 + m;

    // --- A operand: permuted W_hh row m (type = m&3), cols K = 2*hi, 2*hi+1.
    //     Sigmoid rows pre-scaled by 0.5. ---
    const int rowA = (m & 3) * 4 + (m >> 2);         // old gate index of permuted row m
    const float rscale = ((m & 3) == 2) ? 1.0f : 0.5f;
    v2f A;
    A.x = W_hh[rowA * 4 + 2 * hi + 0] * rscale;
    A.y = W_hh[rowA * 4 + 2 * hi + 1] * rscale;

    // --- per-lane x-gate constants for C: permuted row r = v + 8*hi (type = v&3) ---
    float wih[8], bias[8];
#pragma unroll
    for (int v = 0; v < 8; ++v) {
        const int r = v + 8 * hi;
        const int g = (r & 3) * 4 + (r >> 2);        // old gate index
        const float cs = ((v & 3) == 2) ? 1.0f : 0.5f;
        wih[v]  = W_ih[g] * cs;
        bias[v] = (b_ih[g] + b_hh[g]) * cs;
    }

    float c0 = 0.f, c1 = 0.f;                        // cell state: dims (2*hi, 2*hi+1)
    float h0 = 0.f, h1 = 0.f;                        // hidden state, same dims
    v2f Bm; Bm.x = 0.f; Bm.y = 0.f;                  // B operand (h == 0 at t=0)

    const float* xrow = x + (size_t)batch * (size_t)T;

    for (int t0 = 0; t0 < T; t0 += 4) {
        const float4 xq = *(const float4*)(xrow + t0);
        __builtin_prefetch(xrow + t0 + 64, 0, 0);    // global_prefetch ahead of stream
        const float xs[4] = {xq.x, xq.y, xq.z, xq.w};
#pragma unroll
        for (int s = 0; s < 4; ++s) {
            const float xt = xs[s];

            // C = x_t * W_ih[row] + (b_ih[row] + b_hh[row])   (pre-scaled rows)
            v8f C;
#pragma unroll
            for (int v = 0; v < 8; ++v) C[v] = fmaf(xt, wih[v], bias[v]);

            // gates^T = W_hh_perm x h^T + C  -- one fp32 WMMA per 16 batch rows
            v8f D = __builtin_amdgcn_wmma_f32_16x16x4_f32(
                false, A, false, Bm, (short)0, C, false, false);

            // slots: 0=i,1=f,2=g(tanh),3=o for dim n0; 4..7 same for dim n1
            const float i0v = sigmoid_of_half(D[0]);
            const float f0v = sigmoid_of_half(D[1]);
            const float g0v = fast_tanh      (D[2]);
            const float o0v = sigmoid_of_half(D[3]);
            const float i1v = sigmoid_of_half(D[4]);
            const float f1v = sigmoid_of_half(D[5]);
            const float g1v = fast_tanh      (D[6]);
            const float o1v = sigmoid_of_half(D[7]);

            c0 = fmaf(f0v, c0, i0v * g0v);
            c1 = fmaf(f1v, c1, i1v * g1v);
            h0 = o0v * fast_tanh(c0);
            h1 = o1v * fast_tanh(c1);

            // next-step B operand: VGPR0 holds K=2*hi, VGPR1 holds K=2*hi+1
            Bm.x = h0;
            Bm.y = h1;
        }
    }

    // assemble full h per batch in the lower half: dims 2,3 live in lane+16
    const float h2v = __shfl_xor(h0, 16, 32);
    const float h3v = __shfl_xor(h1, 16, 32);

    // head: lin1 = h @ W1^T + b1 ; out = lin1 @ Wll^T + bll  (lower half only)
    if (!hi) {
        const float h4[4] = {h0, h1, h2v, h3v};
        float lin1[10];
#pragma unroll
        for (int j = 0; j < 10; ++j) {
            float a = b1[j];
#pragma unroll
            for (int n = 0; n < 4; ++n) a = fmaf(h4[n], W1[j * 4 + n], a);
            lin1[j] = a;
        }
#pragma unroll
        for (int k = 0; k < 7; ++k) {
            float a = bll[k];
#pragma unroll
            for (int j = 0; j < 10; ++j) a = fmaf(lin1[j], Wll[k * 10 + j], a);
            out[(size_t)batch * 7 + k] = a;
        }
    }
}

extern "C" void kernel_launch(void* const* d_in, const int* in_sizes, int n_in,
                              void* d_out, int out_size, void* d_ws, size_t ws_size,
                              hipStream_t stream) {
    const float* x    = (const float*)d_in[0];
    const float* W_ih = (const float*)d_in[1];
    const float* W_hh = (const float*)d_in[2];
    const float* b_ih = (const float*)d_in[3];
    const float* b_hh = (const float*)d_in[4];
    const float* W1   = (const float*)d_in[5];
    const float* b1   = (const float*)d_in[6];
    const float* Wll  = (const float*)d_in[7];
    const float* bll  = (const float*)d_in[8];
    float* out = (float*)d_out;

    const int B = out_size / 7;            // 4096
    const int T = in_sizes[0] / B;         // 1024
    const int nblocks = B / 16;            // one wave32 per 16 batch rows

    lstm_wmma_f32_16x16x4<<<nblocks, 32, 0, stream>>>(
        x, W_ih, W_hh, b_ih, b_hh, W1, b1, Wll, bll, out, T);
}